// MoENoCustomOps_62878321214326
// MI455X (gfx1250) — compile-verified
//
#include <hip/hip_runtime.h>
#include <hip/hip_bf16.h>
#include <math.h>

typedef __attribute__((ext_vector_type(2))) float v2f;
typedef __attribute__((ext_vector_type(8))) float v8f;

#define NEXP   8
#define HDIM   1024
#define IDIM   4096
#define NTOK   2048      /* B*S */
#define CHUNK  128       /* INTER columns processed per block iteration */

static __device__ __forceinline__ v8f wmma_f32(v2f a, v2f b, v8f c) {
    // D = A(16x4 f32) * B(4x16 f32) + C(16x16 f32)
    return __builtin_amdgcn_wmma_f32_16x16x4_f32(
        /*neg_a=*/false, a, /*neg_b=*/false, b,
        /*c_mod=*/(short)0, c, /*reuse_a=*/false, /*reuse_b=*/false);
}

// ---------------------------------------------------------------------------
// Kernel 0: zero output accumulation region + per-expert counters
// ---------------------------------------------------------------------------
__global__ __launch_bounds__(256) void moe_zero(float* __restrict__ out,
                                                int* __restrict__ counts) {
    int idx = blockIdx.x * 256 + threadIdx.x;
    if (idx < NTOK * HDIM) out[idx] = 0.0f;
    if (idx < NEXP) counts[idx] = 0;
}

// ---------------------------------------------------------------------------
// Kernel 1: router. One wave (32 lanes) per token.
//   logits = x @ Wr^T + b ; softmax -> scores ; top-2 -> per-expert lists
// ---------------------------------------------------------------------------
__global__ __launch_bounds__(256) void moe_router(
    const float* __restrict__ x,       // [NTOK, HDIM]
    const float* __restrict__ rw,      // [NEXP, HDIM]
    const float* __restrict__ rb,      // [NEXP]
    float* __restrict__ scores,        // [NTOK, NEXP]
    int* __restrict__ counts,          // [NEXP]
    int* __restrict__ tlist,           // [NEXP, NTOK]
    float* __restrict__ twght)         // [NEXP, NTOK]
{
    const int wave = threadIdx.x >> 5;
    const int lane = threadIdx.x & 31;
    const int token = blockIdx.x * 8 + wave;
    if (token >= NTOK) return;

    const float* xr = x + (size_t)token * HDIM;
    float acc[NEXP];
#pragma unroll
    for (int e = 0; e < NEXP; ++e) acc[e] = 0.0f;

    for (int h = lane; h < HDIM; h += 32) {
        float xv = xr[h];
#pragma unroll
        for (int e = 0; e < NEXP; ++e) acc[e] += xv * rw[e * HDIM + h];
    }
#pragma unroll
    for (int e = 0; e < NEXP; ++e) {
#pragma unroll
        for (int off = 16; off > 0; off >>= 1)
            acc[e] += __shfl_xor(acc[e], off, 32);
    }

    if (lane == 0) {
        float lg[NEXP];
        float mx = -3.4e38f;
#pragma unroll
        for (int e = 0; e < NEXP; ++e) { lg[e] = acc[e] + rb[e]; mx = fmaxf(mx, lg[e]); }
        float sum = 0.0f;
#pragma unroll
        for (int e = 0; e < NEXP; ++e) { lg[e] = expf(lg[e] - mx); sum += lg[e]; }
        float inv = 1.0f / sum;
#pragma unroll
        for (int e = 0; e < NEXP; ++e) {
            lg[e] *= inv;
            scores[(size_t)token * NEXP + e] = lg[e];
        }
        // top-2 (ties -> lowest index, matching jax.lax.top_k)
        int b1 = 0;
#pragma unroll
        for (int e = 1; e < NEXP; ++e) if (lg[e] > lg[b1]) b1 = e;
        int b2 = (b1 == 0) ? 1 : 0;
#pragma unroll
        for (int e = 0; e < NEXP; ++e) if (e != b1 && lg[e] > lg[b2]) b2 = e;

        int p1 = atomicAdd(&counts[b1], 1);
        tlist[b1 * NTOK + p1] = token;
        twght[b1 * NTOK + p1] = lg[b1];
        int p2 = atomicAdd(&counts[b2], 1);
        tlist[b2 * NTOK + p2] = token;
        twght[b2 * NTOK + p2] = lg[b2];
    }
}

// ---------------------------------------------------------------------------
// Kernel 2: fused expert FFN for one (expert, 16-token tile) per block.
// 256 threads = 8 waves. Dynamic LDS: sX[16*HDIM] + sAct[16*CHUNK].
// ---------------------------------------------------------------------------
__global__ __launch_bounds__(256) void moe_ffn(
    const float* __restrict__ x,       // [NTOK, HDIM]
    const float* __restrict__ gup,     // [NEXP, HDIM, 2*IDIM]
    const float* __restrict__ gub,     // [NEXP, 2*IDIM]
    const float* __restrict__ dwn,     // [NEXP, IDIM, HDIM]
    const float* __restrict__ dwb,     // [NEXP, HDIM]
    const int*   __restrict__ counts,  // [NEXP]
    const int*   __restrict__ tlist,   // [NEXP, NTOK]
    const float* __restrict__ twght,   // [NEXP, NTOK]
    float* __restrict__ out)           // [NTOK, HDIM] (atomic accumulation)
{
    const int e    = blockIdx.y;
    const int tile = blockIdx.x;
    const int cnt  = counts[e];
    if (tile * 16 >= cnt) return;

    extern __shared__ float smem[];
    float* sX   = smem;                 // 16 x HDIM
    float* sAct = smem + 16 * HDIM;     // 16 x CHUNK
    __shared__ int   sTok[16];
    __shared__ float sW[16];

    const int tid  = threadIdx.x;
    const int wave = tid >> 5;
    const int lane = tid & 31;
    const int lan16 = lane & 15;
    const int khalf2 = (lane >> 4) << 1;   // 0 for lanes 0-15, 2 for lanes 16-31

    if (tid < 16) {
        int idx = tile * 16 + tid;
        if (idx < cnt) { sTok[tid] = tlist[e * NTOK + idx]; sW[tid] = twght[e * NTOK + idx]; }
        else           { sTok[tid] = -1;                    sW[tid] = 0.0f; }
    }
    __syncthreads();

    // stage X tile into LDS (rows with no token -> zeros)
    for (int i = tid; i < 16 * HDIM; i += 256) {
        int row = i >> 10;          // HDIM == 1024
        int col = i & (HDIM - 1);
        int tok = sTok[row];
        sX[i] = (tok >= 0) ? x[(size_t)tok * HDIM + col] : 0.0f;
    }
    __syncthreads();

    const float* gup_e = gup + (size_t)e * HDIM * (2 * IDIM);
    const float* dwn_e = dwn + (size_t)e * IDIM * HDIM;

    // persistent down-proj accumulators: this wave owns h in [wave*128, wave*128+128)
    v8f dacc[8];
#pragma unroll
    for (int t = 0; t < 8; ++t) dacc[t] = (v8f){0,0,0,0,0,0,0,0};

    for (int c0 = 0; c0 < IDIM; c0 += CHUNK) {
        // ---- phase 1: gate & up tiles for columns [c0 + wave*16, +16) ----
        v8f gacc = (v8f){0,0,0,0,0,0,0,0};
        v8f uacc = (v8f){0,0,0,0,0,0,0,0};
        const int colg = c0 + wave * 16 + lan16;   // this lane's B column (N)

        for (int kk = 0; kk < HDIM; kk += 4) {
            const int k0 = kk + khalf2;
            v2f a;
            a.x = sX[lan16 * HDIM + k0];
            a.y = sX[lan16 * HDIM + k0 + 1];
            v2f bg, bu;
            const size_t r0 = (size_t)k0 * (2 * IDIM);
            bg.x = gup_e[r0 + colg];
            bg.y = gup_e[r0 + 2 * IDIM + colg];
            bu.x = gup_e[r0 + IDIM + colg];
            bu.y = gup_e[r0 + 2 * IDIM + IDIM + colg];
            gacc = wmma_f32(a, bg, gacc);
            uacc = wmma_f32(a, bu, uacc);
        }

        // bias + SiLU(gate) * up  (elementwise: C layouts of gacc/uacc match)
        const float gb = gub[(size_t)e * (2 * IDIM) + colg];
        const float ub = gub[(size_t)e * (2 * IDIM) + IDIM + colg];
#pragma unroll
        for (int j = 0; j < 8; ++j) {
            float g = gacc[j] + gb;
            float u = uacc[j] + ub;
            float s = g / (1.0f + expf(-g));   // silu
            // C layout: row M = j + 8*(lane>=16), col N = lane&15
            sAct[(j + ((lane >> 4) << 3)) * CHUNK + wave * 16 + lan16] = s * u;
        }
        __syncthreads();

        // ---- phase 2: down-proj accumulation over this CHUNK of K ----
        for (int kk = 0; kk < CHUNK; kk += 4) {
            const int k0 = kk + khalf2;
            v2f a;
            a.x = sAct[lan16 * CHUNK + k0];
            a.y = sAct[lan16 * CHUNK + k0 + 1];
            const size_t krow = (size_t)(c0 + k0) * HDIM;
#pragma unroll
            for (int t = 0; t < 8; ++t) {
                const int hcol = wave * 128 + t * 16 + lan16;
                v2f b;
                b.x = dwn_e[krow + hcol];
                b.y = dwn_e[krow + HDIM + hcol];
                dacc[t] = wmma_f32(a, b, dacc[t]);
            }
        }
        __syncthreads();   // before next chunk overwrites sAct
    }

    // ---- epilogue: + down_bias, * routing weight, atomic scatter ----
#pragma unroll
    for (int t = 0; t < 8; ++t) {
        const int hcol = wave * 128 + t * 16 + lan16;
        const float bias = dwb[(size_t)e * HDIM + hcol];
#pragma unroll
        for (int j = 0; j < 8; ++j) {
            const int row = j + ((lane >> 4) << 3);
            const int tok = sTok[row];
            if (tok >= 0) {
                atomicAdd(&out[(size_t)tok * HDIM + hcol], sW[row] * (dacc[t][j] + bias));
            }
        }
    }
}

// ---------------------------------------------------------------------------
// Host-side launch
// ---------------------------------------------------------------------------
extern "C" void kernel_launch(void* const* d_in, const int* in_sizes, int n_in,
                              void* d_out, int out_size, void* d_ws, size_t ws_size,
                              hipStream_t stream) {
    const float* x   = (const float*)d_in[0];   // hidden_states [2,1024,1024]
    const float* rw  = (const float*)d_in[1];   // router_weight [8,1024]
    const float* rb  = (const float*)d_in[2];   // router_bias [8]
    const float* gup = (const float*)d_in[3];   // gate_up_proj [8,1024,8192]
    const float* gub = (const float*)d_in[4];   // gate_up_bias [8,8192]
    const float* dwn = (const float*)d_in[5];   // down_proj [8,4096,1024]
    const float* dwb = (const float*)d_in[6];   // down_bias [8,1024]

    float* out    = (float*)d_out;                    // [NTOK*HDIM] output
    float* scores = (float*)d_out + (size_t)NTOK * HDIM; // [NTOK*NEXP] scores

    // workspace layout
    char* ws = (char*)d_ws;
    int*   counts = (int*)ws;                                  // 8 ints
    int*   tlist  = (int*)(ws + 64);                           // NEXP*NTOK ints
    float* twght  = (float*)(ws + 64 + NEXP * NTOK * sizeof(int));

    // 0: zero output + counters
    moe_zero<<<(NTOK * HDIM + 255) / 256, 256, 0, stream>>>(out, counts);

    // 1: router (one wave per token)
    moe_router<<<NTOK / 8, 256, 0, stream>>>(x, rw, rb, scores, counts, tlist, twght);

    // 2: expert FFN; worst case one expert takes all 2048 tokens -> 128 tiles
    dim3 grid(128, NEXP);
    size_t lds = (size_t)(16 * HDIM + 16 * CHUNK) * sizeof(float);  // 72 KB
    moe_ffn<<<grid, 256, lds, stream>>>(x, gup, gub, dwn, dwb,
                                        counts, tlist, twght, out);
}